// DefocusMapGenerator_69346541961498
// MI455X (gfx1250) — compile-verified
//
#include <hip/hip_runtime.h>

#define HH 512
#define WW 512
#define NB 4
#define NPIX (NB * HH * WW)   // 4*512*512 = 1048576, == 4096 blocks * 256 threads
#define LAMW 100.0f
#define EPSM 1e-5f
#define NTILE 4096            // NB * (512/16) * (512/16)
#define WAVES 4

// padded LDS tile geometry: 18 rows x 20 cols (cols 18,19 zero), hb 20 rows x 16
#define XPW 20
#define XPN (18 * XPW)        // 360
#define HBN (20 * 16)         // 320
#define PLN 320               // per-wave pattern store: 32 lanes x 5 x v2f

typedef float v2f __attribute__((ext_vector_type(2)));
typedef float v8f __attribute__((ext_vector_type(8)));

// D(16x16,f32) = A(16x4,f32) x B(4x16,f32) + C   -- CDNA5 V_WMMA_F32_16X16X4_F32
static __device__ __forceinline__ v8f wmma4(v2f a, v2f b, v8f c) {
  return __builtin_amdgcn_wmma_f32_16x16x4_f32(false, a, false, b, (short)0, c,
                                               false, false);
}

static __device__ __forceinline__ float nwin(int y, int x) {
  int ylo = y > 0 ? y - 1 : 0, yhi = y < HH - 1 ? y + 1 : HH - 1;
  int xlo = x > 0 ? x - 1 : 0, xhi = x < WW - 1 ? x + 1 : WW - 1;
  return (float)((yhi - ylo + 1) * (xhi - xlo + 1));
}

// ------------------------- one-time setup kernels (VALU, L2-resident) ------

__global__ void k_gray(const float* __restrict__ img, float* __restrict__ g) {
  int i = blockIdx.x * 256 + threadIdx.x;
  int b = i / (HH * WW), yx = i - b * HH * WW;
  const float* p = img + (size_t)b * 3 * HH * WW + yx;
  g[i] = (p[0] + p[HH * WW] + p[2 * HH * WW]) * (1.0f / 3.0f);
}

__global__ void k_blur(const float* __restrict__ src, float* __restrict__ dst,
                       int horiz) {
  int i = blockIdx.x * 256 + threadIdx.x;
  int b = i / (HH * WW), yx = i - b * HH * WW;
  int y = yx / WW, x = yx - y * WW;
  float w[9], s = 0.f;
#pragma unroll
  for (int t = 0; t < 9; ++t) {
    float d = (float)(t - 4);
    w[t] = __expf(-0.5f * d * d);
    s += w[t];
  }
  float inv = 1.0f / s, acc = 0.f;
  const float* pb = src + (size_t)b * HH * WW;
#pragma unroll
  for (int t = 0; t < 9; ++t) {
    int yy = y, xx = x;
    if (horiz) xx = x + t - 4; else yy = y + t - 4;
    if (yy >= 0 && yy < HH && xx >= 0 && xx < WW)
      acc += w[t] * inv * pb[yy * WW + xx];
  }
  dst[i] = acc;
}

static __device__ __forceinline__ void sobel(const float* __restrict__ pb,
                                             int y, int x, float& gx, float& gy) {
  const float kx[3][3] = {{-1.f, 0.f, 1.f}, {-2.f, 0.f, 2.f}, {-1.f, 0.f, 1.f}};
  gx = 0.f; gy = 0.f;
#pragma unroll
  for (int dy = 0; dy < 3; ++dy)
#pragma unroll
    for (int dx = 0; dx < 3; ++dx) {
      int yy = y + dy - 1, xx = x + dx - 1;
      if (yy >= 0 && yy < HH && xx >= 0 && xx < WW) {
        float v = pb[yy * WW + xx];
        gx += v * kx[dy][dx];
        gy += v * kx[dx][dy];
      }
    }
}

__global__ void k_sparse(const float* __restrict__ gray,
                         const float* __restrict__ rebl,
                         float* __restrict__ sparse, float* __restrict__ mask,
                         float* __restrict__ x0) {
  int i = blockIdx.x * 256 + threadIdx.x;
  int b = i / (HH * WW), yx = i - b * HH * WW;
  int y = yx / WW, x = yx - y * WW;
  const float* gb = gray + (size_t)b * HH * WW;
  const float* rb = rebl + (size_t)b * HH * WW;
  float gx, gy;
  sobel(gb, y, x, gx, gy);
  float mag = sqrtf(gx * gx + gy * gy + 1e-12f);
  float edge = mag > 0.05f ? 1.f : 0.f;
  sobel(rb, y, x, gx, gy);
  float magb = sqrtf(gx * gx + gy * gy + 1e-12f);
  float R = mag / (magb + 1e-8f);
  float sig = 1.0f / sqrtf(fmaxf(R * R - 1.f, 1e-6f));
  float sp = fminf(fmaxf(sig, 0.f), 5.f) * edge;
  sparse[i] = sp;
  mask[i] = edge;
  x0[i] = edge * sp;  // x0 = mask * sparse, written straight into d_out
}

// window means + inverse regularized covariance, interleaved 12 floats/pixel:
// coef[i] = { i00,i01,i02,i11 | i12,i22,m0,m1 | m2,0,0,0 }  (3 x float4)
__global__ void k_moments(const float* __restrict__ img,
                          float4* __restrict__ coef) {
  int i = blockIdx.x * 256 + threadIdx.x;
  int b = i / (HH * WW), yx = i - b * HH * WW;
  int y = yx / WW, x = yx - y * WW;
  const float* p0 = img + (size_t)(b * 3 + 0) * HH * WW;
  const float* p1 = img + (size_t)(b * 3 + 1) * HH * WW;
  const float* p2 = img + (size_t)(b * 3 + 2) * HH * WW;
  float s0 = 0, s1 = 0, s2 = 0, q00 = 0, q01 = 0, q02 = 0, q11 = 0, q12 = 0,
        q22 = 0;
#pragma unroll
  for (int dy = -1; dy <= 1; ++dy)
#pragma unroll
    for (int dx = -1; dx <= 1; ++dx) {
      int yy = y + dy, xx = x + dx;
      if (yy >= 0 && yy < HH && xx >= 0 && xx < WW) {
        float v0 = p0[yy * WW + xx], v1 = p1[yy * WW + xx],
              v2 = p2[yy * WW + xx];
        s0 += v0; s1 += v1; s2 += v2;
        q00 += v0 * v0; q01 += v0 * v1; q02 += v0 * v2;
        q11 += v1 * v1; q12 += v1 * v2; q22 += v2 * v2;
      }
    }
  float inw = 1.0f / nwin(y, x);
  float m0 = s0 * inw, m1 = s1 * inw, m2 = s2 * inw;
  float reg = EPSM * inw;
  float a = q00 * inw - m0 * m0 + reg;
  float bq = q01 * inw - m0 * m1;
  float c = q02 * inw - m0 * m2;
  float d = q11 * inw - m1 * m1 + reg;
  float e = q12 * inw - m1 * m2;
  float f = q22 * inw - m2 * m2 + reg;
  float c00 = d * f - e * e, c01 = c * e - bq * f, c02 = bq * e - c * d;
  float det = a * c00 + bq * c01 + c * c02;
  float id = 1.0f / det;
  size_t o = (size_t)i * 3;
  coef[o + 0] = make_float4(c00 * id, c01 * id, c02 * id, (a * f - c * c) * id);
  coef[o + 1] = make_float4((bq * c - a * e) * id, (a * d - bq * bq) * id, m0, m1);
  coef[o + 2] = make_float4(m2, 0.f, 0.f, 0.f);
}

// ------------------------- WMMA 3x3 box filter on a 16x16 tile -------------
// xp: LDS 18x20 halo tile (xp[r][t] = X[r-1][t-1]; cols 18,19 and OOB zero)
// hb: LDS 20x16 scratch (row t = horizontal 3-sum of image row t-1; rows 18,19 zero)
// pl: LDS per-wave pattern store: lane*10 + 2c holds the v2f banded-matrix chunk
//     (identical for horizontal-B and vertical-A since row m == col n == lane&15).
// Horizontal: D[m][n] = sum_t xp[m+1][t] * S[t][n], S[t][n]=1 iff t in {n,n+1,n+2}
// Vertical:   O[m][n] = sum_t R[m][t] * hb[t][n], R[m][t]=1 iff t in {m,m+1,m+2}
static __device__ v8f boxsum_tile(const float* xp, float* hb, const float* pl,
                                  int lane) {
  int n = lane & 15;
  int kb = (lane < 16) ? 0 : 2;  // A/B K-slot base per ISA 16x4 / 4x16 layouts
  const v2f* pv = (const v2f*)(pl + lane * 10);
  // VALU halo rows of horizontal sums: Hp row 0 (image row -1), row 17 (row 16)
  {
    int t = (lane < 16) ? 0 : 17;
    hb[t * 16 + n] = xp[t * XPW + n] + xp[t * XPW + n + 1] + xp[t * XPW + n + 2];
  }
  v8f acc = {0.f, 0.f, 0.f, 0.f, 0.f, 0.f, 0.f, 0.f};
#pragma unroll
  for (int c = 0; c < 5; ++c) {
    int t0 = 4 * c + kb;
    v2f A = *(const v2f*)(xp + (n + 1) * XPW + t0);  // t0 even -> 8B aligned
    acc = wmma4(A, pv[c], acc);
  }
  int mrow = (lane < 16) ? 1 : 9;  // D VGPR i -> rows i / i+8; Hp row = m+1
#pragma unroll
  for (int i = 0; i < 8; ++i) hb[(mrow + i) * 16 + n] = acc[i];
  v8f out = {0.f, 0.f, 0.f, 0.f, 0.f, 0.f, 0.f, 0.f};
#pragma unroll
  for (int c = 0; c < 5; ++c) {
    int t0 = 4 * c + kb;
    v2f Bv;
    Bv.x = hb[t0 * 16 + n];            // rows 18,19 pre-zeroed -> no guard
    Bv.y = hb[(t0 + 1) * 16 + n];
    out = wmma4(pv[c], Bv, out);
  }
  return out;
}

static __device__ __forceinline__ void make_pat(float* pl, int lane) {
  int n = lane & 15;
  int kb = (lane < 16) ? 0 : 2;
  v2f* pv = (v2f*)(pl + lane * 10);
#pragma unroll
  for (int c = 0; c < 5; ++c) {
    int t0 = 4 * c + kb;
    v2f p;
    p.x = (t0 >= n && t0 <= n + 2) ? 1.f : 0.f;
    p.y = (t0 + 1 >= n && t0 + 1 <= n + 2) ? 1.f : 0.f;
    pv[c] = p;
  }
}

static __device__ void load_tile(const float* __restrict__ plane, int y0,
                                 int x0, float* xp, int lane) {
  for (int i = lane; i < XPN; i += 32) {
    int r = i / XPW, t = i - r * XPW;
    int yy = y0 - 1 + r, xx = x0 - 1 + t;
    float v = 0.f;
    if (t < 18 && yy >= 0 && yy < HH && xx >= 0 && xx < WW)
      v = plane[yy * WW + xx];
    xp[i] = v;
  }
}

static __device__ void load_tile_prod(const float* __restrict__ plane,
                                      const float* pp, int y0, int x0,
                                      float* xp, int lane) {
  for (int i = lane; i < XPN; i += 32) {
    int r = i / XPW, t = i - r * XPW;
    int yy = y0 - 1 + r, xx = x0 - 1 + t;
    float v = 0.f;
    if (t < 18 && yy >= 0 && yy < HH && xx >= 0 && xx < WW)
      v = plane[yy * WW + xx] * pp[i];
    xp[i] = v;
  }
}

// p -> a (3 planes), b (1 plane):   pbar, Ipm via WMMA boxsums, then invSig matvec
__global__ __launch_bounds__(128) void k_ab(const float* __restrict__ img,
                                            const float* __restrict__ p,
                                            const float4* __restrict__ coef,
                                            float* __restrict__ av,
                                            float* __restrict__ bv) {
  __shared__ float lds[WAVES * (XPN + XPN + HBN + PLN)];
  int wave = threadIdx.x >> 5, lane = threadIdx.x & 31;
  float* pp = lds + wave * (XPN + XPN + HBN + PLN);
  float* xp = pp + XPN;
  float* hb = xp + XPN;
  float* pl = hb + HBN;
  int tile = blockIdx.x * WAVES + wave;
  int b = tile >> 10, ty = (tile >> 5) & 31, tx = tile & 31;
  int y0 = ty * 16, x0 = tx * 16;
  hb[288 + lane] = 0.f;  // zero pad rows 18,19 (32 entries) once
  make_pat(pl, lane);

  int n = lane & 15, mb = (lane < 16) ? 0 : 8;
  size_t idx0 = ((size_t)b * HH + y0 + mb) * WW + x0 + n;
  __builtin_prefetch((const float*)coef + idx0 * 12, 0, 0);
  __builtin_prefetch((const float*)coef + (idx0 + 4 * WW) * 12, 0, 0);

  load_tile(p + (size_t)b * HH * WW, y0, x0, pp, lane);
  v8f pbox = boxsum_tile(pp, hb, pl, lane);
  v8f ib0, ib1, ib2;
  load_tile_prod(img + (size_t)(b * 3 + 0) * HH * WW, pp, y0, x0, xp, lane);
  ib0 = boxsum_tile(xp, hb, pl, lane);
  load_tile_prod(img + (size_t)(b * 3 + 1) * HH * WW, pp, y0, x0, xp, lane);
  ib1 = boxsum_tile(xp, hb, pl, lane);
  load_tile_prod(img + (size_t)(b * 3 + 2) * HH * WW, pp, y0, x0, xp, lane);
  ib2 = boxsum_tile(xp, hb, pl, lane);

#pragma unroll
  for (int i = 0; i < 8; ++i) {
    int y = y0 + mb + i, x = x0 + n;
    size_t idx = ((size_t)b * HH + y) * WW + x;
    float inw = 1.0f / nwin(y, x);
    float pbar = pbox[i] * inw;
    float4 q0 = coef[idx * 3 + 0];  // i00,i01,i02,i11
    float4 q1 = coef[idx * 3 + 1];  // i12,i22,m0,m1
    float4 q2 = coef[idx * 3 + 2];  // m2,-,-,-
    float m0 = q1.z, m1 = q1.w, m2 = q2.x;
    float c0 = ib0[i] * inw - m0 * pbar;
    float c1 = ib1[i] * inw - m1 * pbar;
    float c2 = ib2[i] * inw - m2 * pbar;
    float a0 = q0.x * c0 + q0.y * c1 + q0.z * c2;
    float a1 = q0.y * c0 + q0.w * c1 + q1.x * c2;
    float a2 = q0.z * c0 + q1.x * c1 + q1.y * c2;
    float bb = pbar - (a0 * m0 + a1 * m1 + a2 * m2);
    av[idx] = a0;
    av[NPIX + idx] = a1;
    av[2 * NPIX + idx] = a2;
    bv[idx] = bb;
  }
}

// a,b -> Amv(p) = Nw*(p - q) + LAM*mask*p   (mode 1: r = p0 = LAM*mask*sparse - Amv)
__global__ __launch_bounds__(128) void k_apply(
    const float* __restrict__ img, const float* __restrict__ p,
    const float* __restrict__ av, const float* __restrict__ bv,
    const float* __restrict__ mask, const float* __restrict__ sparse,
    float* __restrict__ Ap, float* __restrict__ rout, float* __restrict__ pout,
    int mode) {
  __shared__ float lds[WAVES * (XPN + HBN + PLN)];
  int wave = threadIdx.x >> 5, lane = threadIdx.x & 31;
  float* xp = lds + wave * (XPN + HBN + PLN);
  float* hb = xp + XPN;
  float* pl = hb + HBN;
  int tile = blockIdx.x * WAVES + wave;
  int b = tile >> 10, ty = (tile >> 5) & 31, tx = tile & 31;
  int y0 = ty * 16, x0 = tx * 16;
  size_t bof = (size_t)b * HH * WW;
  hb[288 + lane] = 0.f;  // zero pad rows 18,19 once
  make_pat(pl, lane);

  int n = lane & 15, mb = (lane < 16) ? 0 : 8;
  size_t pidx0 = ((size_t)(b * 3) * HH + y0 + mb) * WW + x0 + n;
  __builtin_prefetch(img + pidx0, 0, 0);
  __builtin_prefetch(img + pidx0 + (size_t)HH * WW, 0, 0);
  __builtin_prefetch(img + pidx0 + 2 * (size_t)HH * WW, 0, 0);
  __builtin_prefetch(mask + bof + (size_t)(y0 + mb) * WW + x0 + n, 0, 0);

  load_tile(av + bof, y0, x0, xp, lane);
  v8f s0 = boxsum_tile(xp, hb, pl, lane);
  load_tile(av + NPIX + bof, y0, x0, xp, lane);
  v8f s1 = boxsum_tile(xp, hb, pl, lane);
  load_tile(av + 2 * NPIX + bof, y0, x0, xp, lane);
  v8f s2 = boxsum_tile(xp, hb, pl, lane);
  load_tile(bv + bof, y0, x0, xp, lane);
  v8f s3 = boxsum_tile(xp, hb, pl, lane);

#pragma unroll
  for (int i = 0; i < 8; ++i) {
    int y = y0 + mb + i, x = x0 + n;
    size_t idx = ((size_t)b * HH + y) * WW + x;
    float nw = nwin(y, x);
    float I0 = img[((size_t)(b * 3 + 0) * HH + y) * WW + x];
    float I1 = img[((size_t)(b * 3 + 1) * HH + y) * WW + x];
    float I2 = img[((size_t)(b * 3 + 2) * HH + y) * WW + x];
    float q = (s0[i] * I0 + s1[i] * I1 + s2[i] * I2 + s3[i]) / nw;
    float pv = p[idx];
    float amv = nw * (pv - q) + LAMW * mask[idx] * pv;
    if (mode == 0) {
      Ap[idx] = amv;
    } else {
      float rv = LAMW * mask[idx] * sparse[idx] - amv;
      rout[idx] = rv;
      pout[idx] = rv;
    }
  }
}

// ------------------------- deterministic reductions + CG scalar kernels ----

__global__ void k_dot_part(const float* __restrict__ u,
                           const float* __restrict__ v,
                           float* __restrict__ part) {
  __shared__ float sm[256];
  int i = blockIdx.x * 256 + threadIdx.x;
  sm[threadIdx.x] = u[i] * v[i];
  __syncthreads();
  for (int s = 128; s > 0; s >>= 1) {
    if (threadIdx.x < s) sm[threadIdx.x] += sm[threadIdx.x + s];
    __syncthreads();
  }
  if (threadIdx.x == 0) part[blockIdx.x] = sm[0];
}

__global__ void k_dot_fin(const float* __restrict__ part,
                          float* __restrict__ out) {
  __shared__ float sm[256];
  float acc = 0.f;
  for (int k = threadIdx.x; k < 4096; k += 256) acc += part[k];
  sm[threadIdx.x] = acc;
  __syncthreads();
  for (int s = 128; s > 0; s >>= 1) {
    if (threadIdx.x < s) sm[threadIdx.x] += sm[threadIdx.x + s];
    __syncthreads();
  }
  if (threadIdx.x == 0) *out = sm[0];
}

__global__ void k_xr(float* __restrict__ x, float* __restrict__ r,
                     const float* __restrict__ p, const float* __restrict__ Ap,
                     const float* __restrict__ scal) {
  int i = blockIdx.x * 256 + threadIdx.x;
  float alpha = scal[0] / (scal[1] + 1e-12f);
  x[i] += alpha * p[i];
  r[i] -= alpha * Ap[i];
}

__global__ void k_pup(float* __restrict__ p, const float* __restrict__ r,
                      const float* __restrict__ scal) {
  int i = blockIdx.x * 256 + threadIdx.x;
  float beta = scal[2] / (scal[0] + 1e-12f);
  p[i] = r[i] + beta * p[i];
}

__global__ void k_rs(float* scal) { scal[0] = scal[2]; }

// ------------------------- launcher ----------------------------------------

extern "C" void kernel_launch(void* const* d_in, const int* in_sizes, int n_in,
                              void* d_out, int out_size, void* d_ws,
                              size_t ws_size, hipStream_t stream) {
  (void)in_sizes; (void)n_in; (void)out_size; (void)ws_size;
  const float* img = (const float*)d_in[0];
  float* x = (float*)d_out;  // CG solution lives directly in d_out
  float* ws = (float*)d_ws;

  const size_t N = NPIX;
  float* scal = ws;                 // [0]=rs, [1]=pAp, [2]=rs2
  float* part = ws + 16;            // 4096 block partials
  float* base = ws + 8192;
  float* gray = base + 0 * N;  float* p  = gray;   // reuse after setup
  float* tmp  = base + 1 * N;  float* r  = tmp;
  float* rebl = base + 2 * N;  float* Ap = rebl;
  float* sparse = base + 3 * N;
  float* mask   = base + 4 * N;
  float* coef   = base + 5 * N;     // 12N interleaved (invSig 6 + mu 3 + pad 3)
  float* av     = base + 17 * N;    // 3N
  float* bv     = base + 20 * N;    // 1N  (total 8192 + 21N floats ~ 88 MB)

  dim3 eb(256), eg(N / 256);        // element kernels: 4096 x 256
  dim3 tb(128), tg(NTILE / WAVES);  // tile kernels: 1024 blocks x 4 waves

  k_gray<<<eg, eb, 0, stream>>>(img, gray);
  k_blur<<<eg, eb, 0, stream>>>(gray, tmp, 1);
  k_blur<<<eg, eb, 0, stream>>>(tmp, rebl, 0);
  k_sparse<<<eg, eb, 0, stream>>>(gray, rebl, sparse, mask, x);
  k_moments<<<eg, eb, 0, stream>>>(img, (float4*)coef);

  // r0 = LAM*mask*sparse - Amv(x0);  p0 = r0;  rs = dot(r0,r0)
  k_ab<<<tg, tb, 0, stream>>>(img, x, (const float4*)coef, av, bv);
  k_apply<<<tg, tb, 0, stream>>>(img, x, av, bv, mask, sparse, Ap, r, p, 1);
  k_dot_part<<<eg, eb, 0, stream>>>(r, r, part);
  k_dot_fin<<<1, 256, 0, stream>>>(part, scal + 0);

  for (int it = 0; it < 15; ++it) {
    k_ab<<<tg, tb, 0, stream>>>(img, p, (const float4*)coef, av, bv);
    k_apply<<<tg, tb, 0, stream>>>(img, p, av, bv, mask, sparse, Ap, r, p, 0);
    k_dot_part<<<eg, eb, 0, stream>>>(p, Ap, part);
    k_dot_fin<<<1, 256, 0, stream>>>(part, scal + 1);
    k_xr<<<eg, eb, 0, stream>>>(x, r, p, Ap, scal);
    k_dot_part<<<eg, eb, 0, stream>>>(r, r, part);
    k_dot_fin<<<1, 256, 0, stream>>>(part, scal + 2);
    k_pup<<<eg, eb, 0, stream>>>(p, r, scal);
    k_rs<<<1, 1, 0, stream>>>(scal);
  }
}